// MultiHeadSelfAttention_45311904973221
// MI455X (gfx1250) — compile-verified
//
#include <hip/hip_runtime.h>
#include <math.h>
#include <stdint.h>

// ---------------------------------------------------------------------------
// MI455X (gfx1250) fused multi-head self-attention, fp32 via V_WMMA_F32_16X16X4_F32
// Tile staging via CDNA5 async copy (GLOBAL_LOAD_ASYNC_TO_LDS_B128, ASYNCcnt),
// ping-pong double-buffered in the GEMMs so copies overlap the WMMA stream.
// ---------------------------------------------------------------------------

typedef __attribute__((ext_vector_type(2))) float v2f;
typedef __attribute__((ext_vector_type(8))) float v8f;

#define D_MODEL   1024
#define NUM_HEADS 16
#define HEAD_DIM  64
#define BATCH     4
#define SEQ       2048
#define MROWS     (BATCH * SEQ)   // 8192

static __device__ __forceinline__ v8f wmma4(v2f a, v2f b, v8f c) {
  // D = A(16x4) * B(4x16) + C(16x16), all fp32.
  return __builtin_amdgcn_wmma_f32_16x16x4_f32(false, a, false, b, (short)0, c,
                                               false, false);
}

// Async copy 16B/lane: global -> LDS, bypassing VGPRs (tracked by ASYNCcnt).
// VDST VGPR holds the LDS byte address; flat-LDS aperture keeps the LDS
// offset in addr[31:0], so truncating the generic pointer is correct.
static __device__ __forceinline__ void async_copy16(void* lds_ptr,
                                                    const void* gptr) {
  uint32_t lds = (uint32_t)(uintptr_t)lds_ptr;
  asm volatile("global_load_async_to_lds_b128 %0, %1, off"
               :
               : "v"(lds), "v"(gptr)
               : "memory");
}

static __device__ __forceinline__ void wait_async0() {
  asm volatile("s_wait_asynccnt 0" ::: "memory");
}

// ---------------------------------------------------------------------------
// GEMM: C[M=8192, N=1024] = A[8192,1024] * B[1024,1024] + bias
// AMODE 0: A row-major.  AMODE 1: A gathered from [B,H,N,64] head layout.
// OMODE 0: C row-major.  OMODE 1: C scattered to  [B,H,N,64] head layout.
// Block tile 128x128, 8 waves (4m x 2n), each wave 32x64 (2x4 WMMA accums).
// Ping-pong LDS buffers: async copies for tile i+1 run under tile i's WMMAs.
// ---------------------------------------------------------------------------
template <int AMODE, int OMODE>
__global__ __launch_bounds__(256) void gemm128x128(const float* __restrict__ A,
                                                   const float* __restrict__ Bm,
                                                   const float* __restrict__ bias,
                                                   float* __restrict__ C) {
  constexpr int BM = 128, BN = 128, BK = 16;
  constexpr int LDA = 20;   // bank(20*l+k) distinct; rows 16B-aligned (80B)
  constexpr int LDB = 144;  // bank(16k+col); k,k+2 rows 32 banks apart; 576B rows
  __shared__ float Asm[2][BM * LDA];
  __shared__ float Bsm[2][BK * LDB];

  const int tid  = threadIdx.x;
  const int lane = tid & 31, wave = tid >> 5;
  const int mw = wave >> 1, nw = wave & 1;  // 4x2 wave grid
  const int l15 = lane & 15, hi = lane >> 4;
  const int mBase = blockIdx.x * BM;
  const int nBase = blockIdx.y * BN;

  // Issue the async copies for one 128x16 A tile + 16x128 B tile.
  auto stageTile = [&](int buf, int ko) {
#pragma unroll
    for (int i = 0; i < 2; ++i) {  // A: 512 xb128, 2 per thread
      int idx = tid * 2 + i;
      int r = idx >> 2, c4 = (idx & 3) * 4;
      int grow = mBase + r, gk = ko + c4;
      const float* src;
      if (AMODE == 0) {
        src = A + (size_t)grow * D_MODEL + gk;
      } else {
        int b = grow >> 11, n = grow & (SEQ - 1);
        int h = gk >> 6, dd = gk & 63;  // 4 consecutive gk never cross a head
        src = A + ((size_t)((b * NUM_HEADS + h) * SEQ + n)) * HEAD_DIM + dd;
      }
      async_copy16(&Asm[buf][r * LDA + c4], src);
    }
#pragma unroll
    for (int i = 0; i < 2; ++i) {  // B: 512 xb128, 2 per thread
      int idx = tid * 2 + i;
      int r = idx >> 5, c4 = (idx & 31) * 4;
      async_copy16(&Bsm[buf][r * LDB + c4],
                   &Bm[(size_t)(ko + r) * D_MODEL + nBase + c4]);
    }
  };

  v8f acc[2][4] = {};
  stageTile(0, 0);

  constexpr int NIT = D_MODEL / BK;  // 64
  for (int it = 0; it < NIT; ++it) {
    const int buf = it & 1;
    wait_async0();     // tile `it` (issued last iteration) has landed in LDS
    __syncthreads();   // ... for every wave in the group
    if (it + 1 < NIT) stageTile(buf ^ 1, (it + 1) * BK);  // overlap next copy

    const float* As = Asm[buf];
    const float* Bs = Bsm[buf];
#pragma unroll
    for (int ks = 0; ks < 4; ++ks) {
      int k = ks * 4 + hi * 2;  // A: lanes 0-15 hold k,k+1; lanes 16-31 k+2,k+3
      v2f aF[2], bF[4];
#pragma unroll
      for (int mi = 0; mi < 2; ++mi) {
        int row = mw * 32 + mi * 16 + l15;
        aF[mi] = *(const v2f*)&As[row * LDA + k];
      }
#pragma unroll
      for (int ni = 0; ni < 4; ++ni) {
        int col = nw * 64 + ni * 16 + l15;
        v2f b;
        b.x = Bs[k * LDB + col];
        b.y = Bs[(k + 1) * LDB + col];
        bF[ni] = b;
      }
#pragma unroll
      for (int mi = 0; mi < 2; ++mi)
#pragma unroll
        for (int ni = 0; ni < 4; ++ni)
          acc[mi][ni] = wmma4(aF[mi], bF[ni], acc[mi][ni]);
    }
  }

  // --- epilogue: +bias, store (C layout: VGPR j = rows j / j+8, col = lane&15)
#pragma unroll
  for (int mi = 0; mi < 2; ++mi) {
#pragma unroll
    for (int ni = 0; ni < 4; ++ni) {
      int col = nBase + nw * 64 + ni * 16 + l15;
      float bv = bias[col];
#pragma unroll
      for (int j = 0; j < 8; ++j) {
        int row = mBase + mw * 32 + mi * 16 + hi * 8 + j;
        float v = acc[mi][ni][j] + bv;
        if (OMODE == 0) {
          C[(size_t)row * D_MODEL + col] = v;
        } else {
          int b = row >> 11, n = row & (SEQ - 1);
          int h = col >> 6, dd = col & 63;
          C[((size_t)((b * NUM_HEADS + h) * SEQ + n)) * HEAD_DIM + dd] = v;
        }
      }
    }
  }
}

// ---------------------------------------------------------------------------
// Flash attention: one block per (b,h, 128-query tile). 8 waves x 16 q-rows.
// Q frags in registers (pre-scaled by 1/8); K/V 64-row tiles async-staged in
// LDS; online softmax with 16-lane shuffle reductions; P staged through
// per-wave LDS in two 32-key halves to convert WMMA C-layout -> A-layout.
// ---------------------------------------------------------------------------
__global__ __launch_bounds__(256) void attn_kernel(const float* __restrict__ Q,
                                                   const float* __restrict__ K,
                                                   const float* __restrict__ V,
                                                   float* __restrict__ O) {
  constexpr int BQ = 128, BC = 64;
  constexpr int LDK = 68;  // K^T frag reads: bank (4n+d); 272B rows (16B mult)
  constexpr int LDV = 80;  // V frag rows k,k+2 -> 32 banks apart; 320B rows
  constexpr int LDP = 34;  // per-wave P region, bank (34l+k) distinct
  __shared__ float Ksm[BC * LDK];
  __shared__ float Vsm[BC * LDV];
  __shared__ float Psm[8 * 16 * LDP];  // 54 KB total static LDS

  const int tid = threadIdx.x, lane = tid & 31, wave = tid >> 5;
  const int l15 = lane & 15, hi = lane >> 4;
  const int bh = blockIdx.x >> 4;     // b*H + h
  const int qblk = blockIdx.x & 15;   // which 128-row query tile
  const size_t base = (size_t)bh * SEQ * HEAD_DIM;
  const int qrow0 = qblk * BQ + wave * 16;

  // Q fragments: 16 k-steps over d=64, scale folded in (1/sqrt(64) = 0.125)
  v2f qf[16];
  {
    const float* qp = Q + base + (size_t)(qrow0 + l15) * HEAD_DIM;
#pragma unroll
    for (int ks = 0; ks < 16; ++ks) {
      int d = ks * 4 + hi * 2;
      v2f t = *(const v2f*)(qp + d);
      t.x *= 0.125f;
      t.y *= 0.125f;
      qf[ks] = t;
    }
  }

  v8f o[4] = {};
  float mrow[8], lrow[8];
#pragma unroll
  for (int j = 0; j < 8; ++j) {
    mrow[j] = -1e30f;
    lrow[j] = 0.0f;
  }
  float* Pw = &Psm[wave * 16 * LDP];

  for (int kb = 0; kb < SEQ / BC; ++kb) {
    // --- async-stage K/V tile: 64x64 each, 4 xb128 per thread each ---
    {
      const float* kg = K + base + (size_t)kb * BC * HEAD_DIM;
      const float* vg = V + base + (size_t)kb * BC * HEAD_DIM;
#pragma unroll
      for (int i = 0; i < 4; ++i) {
        int idx = tid + i * 256;  // 0..1023
        int r = idx >> 4, c4 = (idx & 15) * 4;
        async_copy16(&Ksm[r * LDK + c4], &kg[r * HEAD_DIM + c4]);
        async_copy16(&Vsm[r * LDV + c4], &vg[r * HEAD_DIM + c4]);
      }
    }
    wait_async0();
    __syncthreads();

    // --- S = (Q/8) * K^T : 4 n-tiles x 16 k-steps of 4 over d ---
    v8f sacc[4] = {};
#pragma unroll
    for (int nt = 0; nt < 4; ++nt) {
      int n = nt * 16 + l15;
#pragma unroll
      for (int ks = 0; ks < 16; ++ks) {
        int d = ks * 4 + hi * 2;
        v2f b = *(const v2f*)&Ksm[n * LDK + d];  // B[d][n] = K[n][d]
        sacc[nt] = wmma4(qf[ks], b, sacc[nt]);
      }
    }

    // --- online softmax (row j lives in VGPR j across 16 lanes of a half) ---
#pragma unroll
    for (int j = 0; j < 8; ++j) {
      float s0 = fmaxf(fmaxf(sacc[0][j], sacc[1][j]),
                       fmaxf(sacc[2][j], sacc[3][j]));
      s0 = fmaxf(s0, __shfl_xor(s0, 1, 32));
      s0 = fmaxf(s0, __shfl_xor(s0, 2, 32));
      s0 = fmaxf(s0, __shfl_xor(s0, 4, 32));
      s0 = fmaxf(s0, __shfl_xor(s0, 8, 32));
      float mnew = fmaxf(mrow[j], s0);
      float corr = __expf(mrow[j] - mnew);
      float rsum = 0.0f;
#pragma unroll
      for (int nt = 0; nt < 4; ++nt) {
        float p = __expf(sacc[nt][j] - mnew);
        sacc[nt][j] = p;
        rsum += p;
      }
      rsum += __shfl_xor(rsum, 1, 32);
      rsum += __shfl_xor(rsum, 2, 32);
      rsum += __shfl_xor(rsum, 4, 32);
      rsum += __shfl_xor(rsum, 8, 32);
      lrow[j] = lrow[j] * corr + rsum;
      mrow[j] = mnew;
#pragma unroll
      for (int nt = 0; nt < 4; ++nt) o[nt][j] *= corr;
    }

    // --- O += P*V, staged through per-wave LDS in two 32-key halves ---
#pragma unroll
    for (int half = 0; half < 2; ++half) {
#pragma unroll
      for (int t = 0; t < 2; ++t) {
        int nt = half * 2 + t;
#pragma unroll
        for (int j = 0; j < 8; ++j) {
          int row = hi * 8 + j;
          int colk = t * 16 + l15;  // key index within half (0..31)
          Pw[row * LDP + colk] = sacc[nt][j];
        }
      }
#pragma unroll
      for (int ks = 0; ks < 8; ++ks) {  // 32 keys / half
        int kk = ks * 4 + hi * 2;
        v2f a = *(const v2f*)&Pw[l15 * LDP + kk];  // A-layout frag of P
#pragma unroll
        for (int nt = 0; nt < 4; ++nt) {
          int col = nt * 16 + l15;
          v2f b;
          b.x = Vsm[(half * 32 + kk) * LDV + col];
          b.y = Vsm[(half * 32 + kk + 1) * LDV + col];
          o[nt] = wmma4(a, b, o[nt]);
        }
      }
    }
    __syncthreads();  // before next tile overwrites K/V
  }

  // --- normalize and store to [B,H,N,64] head layout ---
#pragma unroll
  for (int j = 0; j < 8; ++j) {
    float inv = 1.0f / lrow[j];
#pragma unroll
    for (int nt = 0; nt < 4; ++nt) o[nt][j] *= inv;
  }
#pragma unroll
  for (int nt = 0; nt < 4; ++nt) {
    int col = nt * 16 + l15;
#pragma unroll
    for (int j = 0; j < 8; ++j) {
      int row = qrow0 + hi * 8 + j;
      O[base + (size_t)row * HEAD_DIM + col] = o[nt][j];
    }
  }
}

// ---------------------------------------------------------------------------
extern "C" void kernel_launch(void* const* d_in, const int* in_sizes, int n_in,
                              void* d_out, int out_size, void* d_ws,
                              size_t ws_size, hipStream_t stream) {
  const float* x  = (const float*)d_in[0];
  const float* Wq = (const float*)d_in[1];
  const float* bq = (const float*)d_in[2];
  const float* Wk = (const float*)d_in[3];
  const float* bk = (const float*)d_in[4];
  const float* Wv = (const float*)d_in[5];
  const float* bv = (const float*)d_in[6];
  const float* Wo = (const float*)d_in[7];
  const float* bo = (const float*)d_in[8];
  float* out = (float*)d_out;

  // Workspace: Q,K,V,Oheads each [B,H,N,64] fp32 -> 4 * 33.5 MB = 134 MB
  const size_t MD = (size_t)MROWS * D_MODEL;
  float* Qh = (float*)d_ws;
  float* Kh = Qh + MD;
  float* Vh = Kh + MD;
  float* Oh = Vh + MD;

  dim3 grid(MROWS / 128, D_MODEL / 128);  // 64 x 8 blocks
  gemm128x128<0, 1><<<grid, 256, 0, stream>>>(x, Wq, bq, Qh);
  gemm128x128<0, 1><<<grid, 256, 0, stream>>>(x, Wk, bk, Kh);
  gemm128x128<0, 1><<<grid, 256, 0, stream>>>(x, Wv, bv, Vh);

  attn_kernel<<<BATCH * NUM_HEADS * (SEQ / 128), 256, 0, stream>>>(Qh, Kh, Vh,
                                                                   Oh);

  gemm128x128<1, 0><<<grid, 256, 0, stream>>>(Oh, Wo, bo, out);
}